// HGCN_PyG_79791902425581
// MI455X (gfx1250) — compile-verified
//
#include <hip/hip_runtime.h>

// ---------------- problem constants (from reference) ----------------
enum : int { N0 = 50000, N1 = 12500, N2 = 3125 };
enum : int { E0 = 800000, E1 = 200000, E2 = 50000 };
enum : int { D = 128 };

typedef __attribute__((ext_vector_type(2))) float v2f;
typedef __attribute__((ext_vector_type(8))) float v8f;

__device__ __forceinline__ int imin(int a, int b) { return a < b ? a : b; }

// ---------------- zero fill ----------------
__global__ void k_zero(float* __restrict__ p, int n) {
    int t = blockIdx.x * blockDim.x + threadIdx.x;
    if (t < n) p[t] = 0.0f;
}

// ---------------- degree: count targets ----------------
__global__ void k_count(const int* __restrict__ col, int* __restrict__ cnt, int E) {
    int t = blockIdx.x * blockDim.x + threadIdx.x;
    if (t < E) atomicAdd(&cnt[col[t]], 1);
}

// dis[i] = (deg_i)^-1/2 with deg = count + 1 (self loop)
__global__ void k_dis(const int* __restrict__ cnt, float* __restrict__ dis, int n) {
    int t = blockIdx.x * blockDim.x + threadIdx.x;
    if (t < n) dis[t] = rsqrtf((float)(cnt[t] + 1));
}

// ---------------- WMMA fp32 GEMM: Y[M x 128] = X[M x 128] @ W[128 x 128] ----------------
// Block = 256 threads (8 wave32s). Block handles a 16-row strip; wave w owns cols [16w,16w+16).
// K-loop: 32 x V_WMMA_F32_16X16X4_F32 per wave.
__global__ void k_gemm128(const float* __restrict__ X, const float* __restrict__ W,
                          float* __restrict__ Y, int M) {
    __shared__ float As[16][132];  // pad 128->132 to avoid bank conflicts
    const int m0 = blockIdx.x * 16;

    // cooperative load of the 16x128 A-strip (zero-pad tail rows)
    for (int t = threadIdx.x; t < 16 * 128; t += 256) {
        int r = t >> 7, k = t & 127;
        int gr = m0 + r;
        As[r][k] = (gr < M) ? X[(long)gr * D + k] : 0.0f;
    }
    __syncthreads();

    const int wave = threadIdx.x >> 5;
    const int lane = threadIdx.x & 31;
    const int n0   = wave * 16;
    const int half = lane >> 4;   // 0: lanes 0-15, 1: lanes 16-31
    const int l15  = lane & 15;

    v8f c = {};
#pragma unroll
    for (int kk = 0; kk < 128; kk += 4) {
        const int ka = kk + 2 * half;  // this half-wave's K pair
        // A 16x4 frag: lane<16 -> K={kk,kk+1}, lane>=16 -> K={kk+2,kk+3}; M = l15
        v2f a;
        a.x = As[l15][ka];
        a.y = As[l15][ka + 1];
        // B 4x16 frag: same K split; N = n0 + l15
        v2f b;
        b.x = W[(long)ka * D + n0 + l15];
        b.y = W[(long)(ka + 1) * D + n0 + l15];
        c = __builtin_amdgcn_wmma_f32_16x16x4_f32(false, a, false, b, (short)0, c,
                                                  false, false);
    }

    // C/D layout: VGPR r -> M = r (lanes 0-15) / r+8 (lanes 16-31), N = l15
#pragma unroll
    for (int r = 0; r < 8; ++r) {
        int row = m0 + r + 8 * half;
        if (row < M) Y[(long)row * D + n0 + l15] = c[r];
    }
}

// ---------------- edge scatter: acc[col] += dis[row]*dis[col] * H[min(row,Mc-1)] ----------------
// one wave per edge, 4 floats per lane (float4 gather, 4x global_atomic_add_f32)
__global__ void k_scatter(float* __restrict__ acc, const float* __restrict__ H,
                          const int* __restrict__ edges, const float* __restrict__ dis,
                          int E, int Mc) {
    int t = blockIdx.x * blockDim.x + threadIdx.x;
    int e = t >> 5, lane = t & 31;
    if (e >= E) return;
    int r = edges[e];
    int cN = edges[E + e];
    float w = dis[r] * dis[cN];
    int rr = imin(r, Mc - 1);
    float4 v = ((const float4*)(H + (long)rr * D))[lane];
    float* ap = acc + (long)cN * D + lane * 4;
    atomicAdd(ap + 0, w * v.x);
    atomicAdd(ap + 1, w * v.y);
    atomicAdd(ap + 2, w * v.z);
    atomicAdd(ap + 3, w * v.w);
}

// ---------------- finish: out = relu(acc + dis[c]^2 * H[min(c,Mc-1)] + b) ----------------
__global__ void k_finish(float* __restrict__ out, const float* __restrict__ acc,
                         const float* __restrict__ H, const float* __restrict__ dis,
                         const float* __restrict__ b, int n, int Mc) {
    int t = blockIdx.x * blockDim.x + threadIdx.x;
    if (t >= n * D) return;
    int c = t >> 7, j = t & 127;
    float d = dis[c];
    int cc = imin(c, Mc - 1);
    float v = acc[t] + d * d * H[(long)cc * D + j] + b[j];
    out[t] = v > 0.0f ? v : 0.0f;
}

// ---------------- pool: acc[idx[i]] += val[i] * h[i] ----------------
__global__ void k_pool(float* __restrict__ acc, const float* __restrict__ h,
                       const int* __restrict__ idx, const float* __restrict__ val, int nf) {
    int t = blockIdx.x * blockDim.x + threadIdx.x;
    int i = t >> 5, lane = t & 31;
    if (i >= nf) return;
    int cl = idx[i];
    float w = val[i];
    float4 v = ((const float4*)(h + (long)i * D))[lane];
    float* ap = acc + (long)cl * D + lane * 4;
    atomicAdd(ap + 0, w * v.x);
    atomicAdd(ap + 1, w * v.y);
    atomicAdd(ap + 2, w * v.z);
    atomicAdd(ap + 3, w * v.w);
}

// ---------------- unpool: out[i] = val[i] * h[min(idx[i],Mc-1)] + skip[i] ----------------
__global__ void k_unpool(float* __restrict__ out, const float* __restrict__ h,
                         const int* __restrict__ idx, const float* __restrict__ val,
                         const float* __restrict__ skip, int nf, int Mc) {
    int t = blockIdx.x * blockDim.x + threadIdx.x;
    if (t >= nf * D) return;
    int i = t >> 7, j = t & 127;
    int cl = imin(idx[i], Mc - 1);
    out[t] = val[i] * h[(long)cl * D + j] + skip[t];
}

// ---------------- row-wise L2 normalize (wave per row) ----------------
__global__ void k_l2norm(float* __restrict__ out, const float* __restrict__ h, int n) {
    int row  = blockIdx.x * 8 + (threadIdx.x >> 5);
    int lane = threadIdx.x & 31;
    if (row >= n) return;
    float4 v = ((const float4*)(h + (long)row * D))[lane];
    float s = v.x * v.x + v.y * v.y + v.z * v.z + v.w * v.w;
#pragma unroll
    for (int m = 16; m >= 1; m >>= 1) s += __shfl_xor(s, m, 32);
    float inv = 1.0f / fmaxf(sqrtf(s), 1e-12f);
    float4 o;
    o.x = v.x * inv; o.y = v.y * inv; o.z = v.z * inv; o.w = v.w * inv;
    ((float4*)(out + (long)row * D))[lane] = o;
}

// =======================================================================
extern "C" void kernel_launch(void* const* d_in, const int* in_sizes, int n_in,
                              void* d_out, int out_size, void* d_ws, size_t ws_size,
                              hipStream_t stream) {
    const float* x      = (const float*)d_in[0];
    const int*   edges0 = (const int*)d_in[1];   // [2, E0] row-major: [0]=src, [1]=dst
    const int*   edges1 = (const int*)d_in[2];
    const int*   edges2 = (const int*)d_in[3];
    const int*   C0_idx = (const int*)d_in[4];
    const float* C0_val = (const float*)d_in[5];
    const int*   C1_idx = (const int*)d_in[6];
    const float* C1_val = (const float*)d_in[7];
    const float *W_enc0 = (const float*)d_in[8],  *b_enc0 = (const float*)d_in[9];
    const float *W_enc1 = (const float*)d_in[10], *b_enc1 = (const float*)d_in[11];
    const float *W_enc2 = (const float*)d_in[12], *b_enc2 = (const float*)d_in[13];
    const float *W_bot  = (const float*)d_in[14], *b_bot  = (const float*)d_in[15];
    const float *W_dec0 = (const float*)d_in[16], *b_dec0 = (const float*)d_in[17];
    const float *W_dec1 = (const float*)d_in[18], *b_dec1 = (const float*)d_in[19];
    const float *W_dec2 = (const float*)d_in[20], *b_dec2 = (const float*)d_in[21];
    float* out = (float*)d_out;

    // ---- workspace carve-up (floats) ----
    float* ws = (float*)d_ws;
    size_t off = 0;
    float* dis0 = ws + off; off += N0;
    float* dis1 = ws + off; off += N1;
    float* dis2 = ws + off; off += N2;
    int*   cnt  = (int*)(ws + off); off += N0;
    float* t0 = ws + off; off += (size_t)N0 * D;
    float* t1 = ws + off; off += (size_t)N0 * D;
    float* e0 = ws + off; off += (size_t)N0 * D;
    float* e1 = ws + off; off += (size_t)N1 * D;
    (void)off; (void)ws_size; (void)n_in; (void)in_sizes; (void)out_size;

    const int B = 256;
    auto cdiv = [](long a, long b) { return (int)((a + b - 1) / b); };

#define ZERO(p, n)            k_zero<<<cdiv((long)(n), B), B, 0, stream>>>((float*)(p), (int)(n))
#define GEMM(X, W, Y, M)      k_gemm128<<<cdiv((long)(M), 16), B, 0, stream>>>((X), (W), (Y), (M))
#define SCAT(acc, H, E_, d, Ecnt, Mc) \
    k_scatter<<<cdiv((long)(Ecnt) * 32, B), B, 0, stream>>>((acc), (H), (E_), (d), (Ecnt), (Mc))
#define FIN(o, a, H, d, b, n, Mc) \
    k_finish<<<cdiv((long)(n) * D, B), B, 0, stream>>>((o), (a), (H), (d), (b), (n), (Mc))
#define POOL(acc, h, idx, val, nf) \
    k_pool<<<cdiv((long)(nf) * 32, B), B, 0, stream>>>((acc), (h), (idx), (val), (nf))
#define UNPOOL(o, h, idx, val, sk, nf, Mc) \
    k_unpool<<<cdiv((long)(nf) * D, B), B, 0, stream>>>((o), (h), (idx), (val), (sk), (nf), (Mc))

    // ---- degree norms per level ----
    ZERO(cnt, N0); k_count<<<cdiv(E0, B), B, 0, stream>>>(edges0 + E0, cnt, E0);
    k_dis<<<cdiv(N0, B), B, 0, stream>>>(cnt, dis0, N0);
    ZERO(cnt, N1); k_count<<<cdiv(E1, B), B, 0, stream>>>(edges1 + E1, cnt, E1);
    k_dis<<<cdiv(N1, B), B, 0, stream>>>(cnt, dis1, N1);
    ZERO(cnt, N2); k_count<<<cdiv(E2, B), B, 0, stream>>>(edges2 + E2, cnt, E2);
    k_dis<<<cdiv(N2, B), B, 0, stream>>>(cnt, dis2, N2);

    // ---- enc0 (n=N0, M=N0) -> e0 ----
    GEMM(x, W_enc0, t0, N0);
    ZERO(t1, (size_t)N0 * D);
    SCAT(t1, t0, edges0, dis0, E0, N0);
    FIN(e0, t1, t0, dis0, b_enc0, N0, N0);

    // ---- pool0: e0 (N0) -> t0 (N1) ----
    ZERO(t0, (size_t)N1 * D);
    POOL(t0, e0, C0_idx, C0_val, N0);

    // ---- enc1 (n=N1, M=N1) -> e1 ----
    GEMM(t0, W_enc1, t1, N1);
    ZERO(t0, (size_t)N1 * D);
    SCAT(t0, t1, edges1, dis1, E1, N1);
    FIN(e1, t0, t1, dis1, b_enc1, N1, N1);

    // ---- pool1: e1 (N1) -> t0 (N2) ----
    ZERO(t0, (size_t)N2 * D);
    POOL(t0, e1, C1_idx, C1_val, N1);

    // ---- enc2 (n=N2, M=N2) -> t0 ----
    GEMM(t0, W_enc2, t1, N2);
    ZERO(t0, (size_t)N2 * D);
    SCAT(t0, t1, edges2, dis2, E2, N2);
    FIN(t0, t0, t1, dis2, b_enc2, N2, N2);

    // ---- bottleneck (n=N2, M=N2) -> t0 ----
    GEMM(t0, W_bot, t1, N2);
    ZERO(t0, (size_t)N2 * D);
    SCAT(t0, t1, edges2, dis2, E2, N2);
    FIN(t0, t0, t1, dis2, b_bot, N2, N2);

    // ---- dec l=2: unpool C1 (N2 -> N1 rows), gcn at n=N2 with M=N1 -> t1 (N2) ----
    UNPOOL(t1, t0, C1_idx, C1_val, e1, N1, N2);
    GEMM(t1, W_dec2, t0, N1);
    ZERO(t1, (size_t)N2 * D);
    SCAT(t1, t0, edges2, dis2, E2, N1);
    FIN(t1, t1, t0, dis2, b_dec2, N2, N1);

    // ---- dec l=1: unpool C0 (N2 rows! clamp idx) -> N0 rows, gcn at n=N1, M=N0 -> t0 (N1) ----
    UNPOOL(t0, t1, C0_idx, C0_val, e0, N0, N2);
    GEMM(t0, W_dec1, t1, N0);
    ZERO(t0, (size_t)N1 * D);
    SCAT(t0, t1, edges1, dis1, E1, N0);
    FIN(t0, t0, t1, dis1, b_dec1, N1, N0);

    // ---- dec l=0: gcn at n=N0 with M=N1 (clamped gathers) -> t0 (N0) ----
    GEMM(t0, W_dec0, t1, N1);
    ZERO(t0, (size_t)N0 * D);
    SCAT(t0, t1, edges0, dis0, E0, N1);
    FIN(t0, t0, t1, dis0, b_dec0, N0, N1);

    // ---- final L2 normalize ----
    k_l2norm<<<cdiv((long)N0, 8), B, 0, stream>>>(out, t0, N0);

#undef ZERO
#undef GEMM
#undef SCAT
#undef FIN
#undef POOL
#undef UNPOOL
}